// SAGEGNNEncoder_36429912605373
// MI455X (gfx1250) — compile-verified
//
#include <hip/hip_runtime.h>
#include <cstdint>

typedef __attribute__((ext_vector_type(2))) float v2f;
typedef __attribute__((ext_vector_type(8))) float v8f;

constexpr int FEAT = 64;

// Native CDNA5 device-scope float atomic max (no CAS loop).
__device__ __forceinline__ void atomic_max_f32_dev(float* addr, float val) {
  asm volatile("global_atomic_max_num_f32 %0, %1, off scope:SCOPE_DEV"
               :: "v"(addr), "v"(val)
               : "memory");
}

// jnp.where(isfinite(v), v, 0): -inf (empty segment), +inf, NaN -> 0, branch-free
__device__ __forceinline__ float fix_finite(float v) {
  return (v > -3.0e38f && v < 3.0e38f) ? v : 0.0f;
}

// Wc = Wr + Ws ; bias = bl + bs   (fuses SAGE root-linear with the skip Linear)
__global__ void __launch_bounds__(256)
prep_weights_kernel(const float* __restrict__ Wr, const float* __restrict__ Ws,
                    const float* __restrict__ bl, const float* __restrict__ bs,
                    float* __restrict__ Wc, float* __restrict__ bias) {
  int tid = blockIdx.x * blockDim.x + threadIdx.x;
  if (tid < FEAT * FEAT) Wc[tid] = Wr[tid] + Ws[tid];
  if (tid < FEAT)        bias[tid] = bl[tid] + bs[tid];
}

__global__ void __launch_bounds__(256)
init_neg_inf_kernel(float4* __restrict__ agg, int n4) {
  int tid = blockIdx.x * blockDim.x + threadIdx.x;
  if (tid < n4) {
    float4 v;
    v.x = v.y = v.z = v.w = -__builtin_inff();
    agg[tid] = v;
  }
}

// 16 threads per edge; coalesced float4 gather of source row, 4 native f32-max atomics
__global__ void __launch_bounds__(256)
scatter_max_kernel(const float4* __restrict__ h4, const int* __restrict__ src,
                   const int* __restrict__ dst, float* __restrict__ agg, int total) {
  int tid = blockIdx.x * blockDim.x + threadIdx.x;
  if (tid >= total) return;
  int e = tid >> 4;
  int q = tid & 15;
  int s = __ldg(src + e);
  int d = __ldg(dst + e);
  float4 v = h4[(size_t)s * (FEAT / 4) + q];
  float* base = agg + (size_t)d * FEAT + q * 4;
  atomic_max_f32_dev(base + 0, v.x);
  atomic_max_f32_dev(base + 1, v.y);
  atomic_max_f32_dev(base + 2, v.z);
  atomic_max_f32_dev(base + 3, v.w);
}

// out = relu( fix(agg) @ Wl^T + h @ Wc^T + bias )
// One wave computes a 16-row x 64-col tile via v_wmma_f32_16x16x4_f32.
// Weights staged in LDS (2 x 16KB) once per 4-wave block.
__global__ void __launch_bounds__(128)
sage_fused_gemm_kernel(const float* __restrict__ agg, const float* __restrict__ h,
                       const float* __restrict__ Wl, const float* __restrict__ Wc,
                       const float* __restrict__ bias, float* __restrict__ out,
                       int ntiles, int N) {
  __shared__ float sWl[FEAT * FEAT];
  __shared__ float sWc[FEAT * FEAT];
  __shared__ float sB[FEAT];

  {
    const float4* gL = (const float4*)Wl;
    const float4* gC = (const float4*)Wc;
    float4* lL = (float4*)sWl;
    float4* lC = (float4*)sWc;
    for (int i = threadIdx.x; i < FEAT * FEAT / 4; i += 128) {
      lL[i] = gL[i];
      lC[i] = gC[i];
    }
    if (threadIdx.x < FEAT) sB[threadIdx.x] = bias[threadIdx.x];
  }
  __syncthreads();

  const int wave = threadIdx.x >> 5;
  const int lane = threadIdx.x & 31;
  const int tile = blockIdx.x * 4 + wave;
  if (tile >= ntiles) return;

  const int row0 = tile * 16;
  const int m16  = lane & 15;   // M (A/C) or N (B/C) position within 16
  const int kh   = lane >> 4;   // K-half for A/B fragments, M-half for C/D

  v8f acc[4];
#pragma unroll
  for (int nt = 0; nt < 4; ++nt) acc[nt] = (v8f)0.0f;

  int r = row0 + m16;
  if (r >= N) r = N - 1;                      // clamp loads on ragged tail
  const float2* agg2 = (const float2*)agg;
  const float2* h2   = (const float2*)h;

#pragma unroll
  for (int ks = 0; ks < 16; ++ks) {           // K = 64 in steps of 4
    const int base_k = ks * 4 + kh * 2;
    const int idx2 = r * (FEAT / 2) + (base_k >> 1);
    float2 av = agg2[idx2];
    float2 hv = h2[idx2];
    v2f aA; aA.x = fix_finite(av.x); aA.y = fix_finite(av.y);
    v2f aH; aH.x = hv.x;             aH.y = hv.y;

#pragma unroll
    for (int nt = 0; nt < 4; ++nt) {
      const int n = nt * 16 + m16;
      float2 blv = *(const float2*)(&sWl[n * FEAT + base_k]);  // B[k][n] = W[n][k]
      float2 bcv = *(const float2*)(&sWc[n * FEAT + base_k]);
      v2f bL; bL.x = blv.x; bL.y = blv.y;
      v2f bC; bC.x = bcv.x; bC.y = bcv.y;
      acc[nt] = __builtin_amdgcn_wmma_f32_16x16x4_f32(
          false, aA, false, bL, (short)0, acc[nt], false, false);
      acc[nt] = __builtin_amdgcn_wmma_f32_16x16x4_f32(
          false, aH, false, bC, (short)0, acc[nt], false, false);
    }
  }

  // Epilogue: +bias, ReLU, store.  C/D layout: VGPR v -> M = v + kh*8, N = m16.
#pragma unroll
  for (int nt = 0; nt < 4; ++nt) {
    const int n = nt * 16 + m16;
    const float bv = sB[n];
#pragma unroll
    for (int v = 0; v < 8; ++v) {
      const int m = row0 + v + kh * 8;
      if (m < N) {
        float val = acc[nt][v] + bv;
        out[(size_t)m * FEAT + n] = val > 0.0f ? val : 0.0f;
      }
    }
  }
}

extern "C" void kernel_launch(void* const* d_in, const int* in_sizes, int n_in,
                              void* d_out, int out_size, void* d_ws, size_t ws_size,
                              hipStream_t stream) {
  const float* x  = (const float*)d_in[0];
  const int*   ei = (const int*)d_in[1];
  const int N = in_sizes[0] / FEAT;
  const int E = in_sizes[1] / 2;
  const int* src = ei;         // edge_index[0]
  const int* dst = ei + E;     // edge_index[1]

  const float *Wl[3], *bl[3], *Wr[3], *Ws[3], *bs[3];
  for (int l = 0; l < 3; ++l) {
    Wl[l] = (const float*)d_in[2 + 5 * l + 0];
    bl[l] = (const float*)d_in[2 + 5 * l + 1];
    Wr[l] = (const float*)d_in[2 + 5 * l + 2];
    Ws[l] = (const float*)d_in[2 + 5 * l + 3];
    bs[l] = (const float*)d_in[2 + 5 * l + 4];
  }

  // Workspace carve-up
  const size_t nodeBytes = (size_t)N * FEAT * sizeof(float);
  char* ws = (char*)d_ws;
  float* agg = (float*)ws; ws += nodeBytes;
  float* hA  = (float*)ws; ws += nodeBytes;
  float* hB  = (float*)ws; ws += nodeBytes;
  float* Wc  = (float*)ws; ws += (size_t)3 * FEAT * FEAT * sizeof(float);
  float* bc  = (float*)ws; ws += (size_t)3 * FEAT * sizeof(float);

  for (int l = 0; l < 3; ++l) {
    prep_weights_kernel<<<(FEAT * FEAT + 255) / 256, 256, 0, stream>>>(
        Wr[l], Ws[l], bl[l], bs[l], Wc + l * FEAT * FEAT, bc + l * FEAT);
  }

  const float* hin = x;
  float* houts[3] = {hA, hB, (float*)d_out};

  const int n4 = N * FEAT / 4;
  const int scatterTotal = E * 16;
  const int ntiles = (N + 15) / 16;

  for (int l = 0; l < 3; ++l) {
    init_neg_inf_kernel<<<(n4 + 255) / 256, 256, 0, stream>>>((float4*)agg, n4);
    scatter_max_kernel<<<(scatterTotal + 255) / 256, 256, 0, stream>>>(
        (const float4*)hin, src, dst, agg, scatterTotal);
    sage_fused_gemm_kernel<<<(ntiles + 3) / 4, 128, 0, stream>>>(
        agg, hin, Wl[l], Wc + l * FEAT * FEAT, bc + l * FEAT, houts[l], ntiles, N);
    hin = houts[l];
  }
}